// Attention1_70841190580880
// MI455X (gfx1250) — compile-verified
//
#include <hip/hip_runtime.h>
#include <hip/hip_bf16.h>
#include <stdint.h>

// ---------------- problem constants ----------------
#define BATCH 8
#define SEQ   2048
#define EDIM  512
#define VOCAB 22
#define VPAD  32
#define INV_TEMP 0.044194173824159216f   // 1/sqrt(512)

typedef float v2f __attribute__((ext_vector_type(2)));
typedef float v4f __attribute__((ext_vector_type(4)));
typedef float v8f __attribute__((ext_vector_type(8)));

// ---------------- workspace byte offsets ----------------
#define OFF_QKV     0           // f32 [3][32][512]            196608 B
#define OFF_S22     196608      // f32 [32][32]                  4096 B
#define OFF_CNT     200704      // i32 [8][32]                   1024 B
#define OFF_CNTALL  201728      // i32 [8][32]                   1024 B
#define OFF_CM      202752      // u8  [8][2048]                16384 B
#define OFF_ROWVAL  219136      // f32 [8][32][24]              24576 B
#define OFF_W22     243712      // f32 [8][32][32]              32768 B
#define OFF_OUT22   276480      // f32 [8][32][512]            524288 B
// total: 800768 B

__device__ __forceinline__ v8f wmma4(v2f a, v2f b, v8f c) {
  // V_WMMA_F32_16X16X4_F32 : full f32 precision, RNE
  return __builtin_amdgcn_wmma_f32_16x16x4_f32(false, a, false, b,
                                               (short)0, c, false, false);
}

// ---------------------------------------------------------------------------
// 1) QKV tables: dst[w][32][512] = pad22(emb) @ W{q,k,v}   (Q scaled by 1/T)
//    grid 3*2*32 blocks, 1 wave each
// ---------------------------------------------------------------------------
__global__ __launch_bounds__(32) void k_qkv(const float* __restrict__ emb,
                                            const float* __restrict__ Wq,
                                            const float* __restrict__ Wk,
                                            const float* __restrict__ Wv,
                                            float* __restrict__ qkv) {
  const int blk   = blockIdx.x;       // 0..191
  const int w     = blk / 64;         // 0=Q 1=K 2=V
  const int rem   = blk % 64;
  const int mtile = rem >> 5;         // 0..1
  const int ntile = rem & 31;         // 0..31
  const float* W  = (w == 0) ? Wq : ((w == 1) ? Wk : Wv);
  float* dst      = qkv + (size_t)w * VPAD * EDIM;

  const int lane = threadIdx.x & 31;
  const int ln   = lane & 15;
  const int hi   = lane >> 4;
  const int koff = hi * 2;
  const int arow = mtile * 16 + ln;   // emb row (pad >=22 with zeros)
  const int col  = ntile * 16 + ln;   // output column

  v8f acc = {};
  for (int k0 = 0; k0 < EDIM; k0 += 16) {
#pragma unroll
    for (int kk = 0; kk < 16; kk += 4) {
      const int k = k0 + kk + koff;
      v2f a, b;
      if (arow < VOCAB) { a.x = emb[arow * EDIM + k]; a.y = emb[arow * EDIM + k + 1]; }
      else              { a.x = 0.f;                  a.y = 0.f; }
      b.x = W[(size_t)k * EDIM + col];
      b.y = W[(size_t)(k + 1) * EDIM + col];
      acc = wmma4(a, b, acc);
    }
  }
  const float scale = (w == 0) ? INV_TEMP : 1.0f;
#pragma unroll
  for (int r = 0; r < 8; ++r) {
    const int orow = mtile * 16 + r + hi * 8;
    dst[(size_t)orow * EDIM + col] = acc[r] * scale;
  }
}

// ---------------------------------------------------------------------------
// 2) S22[32][32] = Qs @ K^T   (1 block, 4 waves, one 16x16 tile each)
// ---------------------------------------------------------------------------
__global__ __launch_bounds__(128) void k_s22(const float* __restrict__ qkv,
                                             float* __restrict__ s22) {
  const int wave  = threadIdx.x >> 5;
  const int mtile = wave >> 1;
  const int ntile = wave & 1;
  const int lane  = threadIdx.x & 31;
  const int ln    = lane & 15;
  const int hi    = lane >> 4;
  const int koff  = hi * 2;
  const float* Q  = qkv;                 // [32][512]
  const float* K  = qkv + VPAD * EDIM;   // [32][512]
  const int arow  = mtile * 16 + ln;
  const int brow  = ntile * 16 + ln;

  v8f acc = {};
  for (int k0 = 0; k0 < EDIM; k0 += 16) {
#pragma unroll
    for (int kk = 0; kk < 16; kk += 4) {
      const int k = k0 + kk + koff;
      v2f a = { Q[arow * EDIM + k], Q[arow * EDIM + k + 1] };
      v2f b = { K[brow * EDIM + k], K[brow * EDIM + k + 1] };
      acc = wmma4(a, b, acc);
    }
  }
#pragma unroll
  for (int r = 0; r < 8; ++r)
    s22[(mtile * 16 + r + hi * 8) * VPAD + ntile * 16 + ln] = acc[r];
}

// ---------------------------------------------------------------------------
// 3) Per-batch class histogram + column class codes (22 == masked out)
// ---------------------------------------------------------------------------
__global__ __launch_bounds__(256) void k_hist(const int* __restrict__ ids,
                                              const int* __restrict__ mask,
                                              int* __restrict__ cnt,
                                              int* __restrict__ cntAll,
                                              unsigned char* __restrict__ cm) {
  __shared__ int sc[VPAD], sa[VPAD];
  const int b = blockIdx.x;
  if (threadIdx.x < VPAD) { sc[threadIdx.x] = 0; sa[threadIdx.x] = 0; }
  __syncthreads();
  for (int m = threadIdx.x; m < SEQ; m += 256) {
    const int t  = ids[b * SEQ + m];
    const int mk = mask[b * SEQ + m];
    atomicAdd(&sa[t], 1);
    if (mk) atomicAdd(&sc[t], 1);
    cm[b * SEQ + m] = (unsigned char)(mk ? t : VOCAB);
  }
  __syncthreads();
  if (threadIdx.x < VPAD) {
    cnt[b * VPAD + threadIdx.x]    = sc[threadIdx.x];
    cntAll[b * VPAD + threadIdx.x] = sa[threadIdx.x];
  }
}

// ---------------------------------------------------------------------------
// 4) Per-batch softmax row tables: rowVal[b][i][0..22] and w22[b][i][0..31]
// ---------------------------------------------------------------------------
__global__ __launch_bounds__(32) void k_rows(const float* __restrict__ s22,
                                             const int* __restrict__ cnt,
                                             const int* __restrict__ cntAll,
                                             float* __restrict__ rowVal,
                                             float* __restrict__ w22) {
  const int b = blockIdx.x;
  const int i = threadIdx.x;
  float* rv = rowVal + ((size_t)b * VPAD + i) * 24;
  float* wr = w22 + ((size_t)b * VPAD + i) * VPAD;
  if (i >= VOCAB) {               // padding rows: zero A-matrix rows
    for (int t = 0; t < VPAD; ++t) wr[t] = 0.f;
    return;
  }
  const int* c  = cnt + b * VPAD;
  const int* ca = cntAll + b * VPAD;
  int total = 0;
  for (int t = 0; t < VOCAB; ++t) total += c[t];

  if (total > 0) {
    float mx = -3.0e38f;
    for (int t = 0; t < VOCAB; ++t)
      if (c[t] > 0) mx = fmaxf(mx, s22[i * VPAD + t]);
    float e[VOCAB];
    float Z = 0.f;
    for (int t = 0; t < VOCAB; ++t) {
      const float ev = (c[t] > 0) ? __expf(s22[i * VPAD + t] - mx) : 0.f;
      e[t] = ev;
      Z += (float)c[t] * ev;
    }
    const float inv = 1.f / Z;
    for (int t = 0; t < VOCAB; ++t) {
      rv[t] = e[t] * inv;                       // prob of an unmasked col of class t
      wr[t] = (float)c[t] * e[t] * inv;         // aggregate weight for V row t
    }
    rv[VOCAB] = 0.f;                            // masked columns -> exp underflows to 0
    for (int t = VOCAB; t < VPAD; ++t) wr[t] = 0.f;
  } else {
    // all columns masked: softmax of constant -1e9 row -> uniform 1/L
    const float u = 1.0f / (float)SEQ;
    for (int t = 0; t <= VOCAB; ++t) rv[t] = u;
    for (int t = 0; t < VOCAB; ++t) wr[t] = (float)ca[t] * u;
    for (int t = VOCAB; t < VPAD; ++t) wr[t] = 0.f;
  }
}

// ---------------------------------------------------------------------------
// 5) out22[b] = w22[b] @ Vemb   (grid 8*2*32 blocks, 1 wave each)
// ---------------------------------------------------------------------------
__global__ __launch_bounds__(32) void k_out22(const float* __restrict__ w22,
                                              const float* __restrict__ qkv,
                                              float* __restrict__ out22) {
  const int blk   = blockIdx.x;           // 0..511
  const int b     = blk >> 6;
  const int rem   = blk & 63;
  const int mtile = rem >> 5;
  const int ntile = rem & 31;
  const float* A  = w22 + (size_t)b * VPAD * VPAD;
  const float* V  = qkv + (size_t)2 * VPAD * EDIM;

  const int lane = threadIdx.x & 31;
  const int ln   = lane & 15;
  const int hi   = lane >> 4;
  const int koff = hi * 2;
  const int arow = mtile * 16 + ln;
  const int col  = ntile * 16 + ln;

  v8f acc = {};
#pragma unroll
  for (int k0 = 0; k0 < VPAD; k0 += 4) {
    const int k = k0 + koff;
    v2f a = { A[arow * VPAD + k], A[arow * VPAD + k + 1] };
    v2f b2 = { V[(size_t)k * EDIM + col], V[(size_t)(k + 1) * EDIM + col] };
    acc = wmma4(a, b2, acc);
  }
  float* D = out22 + (size_t)b * VPAD * EDIM;
#pragma unroll
  for (int r = 0; r < 8; ++r)
    D[(size_t)(mtile * 16 + r + hi * 8) * EDIM + col] = acc[r];
}

// ---------------------------------------------------------------------------
// 6) out[b,l,:] = out22[b][id[b,l]][:]   (NT b128 streaming stores, 33.5 MB)
// ---------------------------------------------------------------------------
__global__ __launch_bounds__(128) void k_wout(const int* __restrict__ ids,
                                              const float* __restrict__ out22,
                                              float* __restrict__ out) {
  const int row = blockIdx.x;              // 0..16383
  const int b   = row >> 11;
  const int l   = row & (SEQ - 1);
  const int i   = ids[b * SEQ + l];
  const v4f* src = (const v4f*)(out22 + ((size_t)b * VPAD + i) * EDIM);
  v4f* dst       = (v4f*)(out + (size_t)row * EDIM);
  const v4f v = src[threadIdx.x];
  __builtin_nontemporal_store(v, &dst[threadIdx.x]);
}

// ---------------------------------------------------------------------------
// 7) attn[b,l,m] = rowVal[b][id_l][cm[b][m]]  (NT streaming stores, 134 MB)
//    one block per (b, 16-row tile of l)
// ---------------------------------------------------------------------------
__global__ __launch_bounds__(256) void k_wattn(const int* __restrict__ ids,
                                               const unsigned char* __restrict__ cm,
                                               const float* __restrict__ rowVal,
                                               float* __restrict__ attn) {
  __shared__ unsigned char scm[SEQ];
  __shared__ float srv[VOCAB * 24];
  __shared__ int sid[16];

  const int b     = blockIdx.x >> 7;
  const int lbase = (blockIdx.x & 127) << 4;

  const uint32_t* cms  = (const uint32_t*)(cm + (size_t)b * SEQ);
  uint32_t* scm32      = (uint32_t*)scm;
  for (int j = threadIdx.x; j < SEQ / 4; j += 256) scm32[j] = cms[j];
  for (int j = threadIdx.x; j < VOCAB * 24; j += 256)
    srv[j] = rowVal[(size_t)b * VPAD * 24 + j];
  if (threadIdx.x < 16) sid[threadIdx.x] = ids[b * SEQ + lbase + threadIdx.x];
  __syncthreads();

  const int mbase = threadIdx.x * 8;
  for (int lr = 0; lr < 16; ++lr) {
    const int i = sid[lr];
    const float* rb = &srv[i * 24];
    float* dst = attn + ((size_t)(b * SEQ + lbase + lr)) * SEQ + mbase;
    v4f va, vb;
    va.x = rb[scm[mbase + 0]];
    va.y = rb[scm[mbase + 1]];
    va.z = rb[scm[mbase + 2]];
    va.w = rb[scm[mbase + 3]];
    vb.x = rb[scm[mbase + 4]];
    vb.y = rb[scm[mbase + 5]];
    vb.z = rb[scm[mbase + 6]];
    vb.w = rb[scm[mbase + 7]];
    __builtin_nontemporal_store(va, (v4f*)dst);
    __builtin_nontemporal_store(vb, (v4f*)(dst + 4));
  }
}

// ---------------------------------------------------------------------------
extern "C" void kernel_launch(void* const* d_in, const int* in_sizes, int n_in,
                              void* d_out, int out_size, void* d_ws, size_t ws_size,
                              hipStream_t stream) {
  const int*   ids  = (const int*)d_in[0];     // [8,1,2048]
  const int*   mask = (const int*)d_in[1];     // [8,1,1,2048]
  const float* emb  = (const float*)d_in[2];   // [22,512]
  const float* Wq   = (const float*)d_in[3];   // [512,512]
  const float* Wk   = (const float*)d_in[4];
  const float* Wv   = (const float*)d_in[5];

  char* ws = (char*)d_ws;
  float* qkv            = (float*)(ws + OFF_QKV);
  float* s22            = (float*)(ws + OFF_S22);
  int*   cnt            = (int*)(ws + OFF_CNT);
  int*   cntAll         = (int*)(ws + OFF_CNTALL);
  unsigned char* cm     = (unsigned char*)(ws + OFF_CM);
  float* rowVal         = (float*)(ws + OFF_ROWVAL);
  float* w22            = (float*)(ws + OFF_W22);
  float* out22          = (float*)(ws + OFF_OUT22);

  float* out  = (float*)d_out;                        // [8,1,2048,512]
  float* attn = out + (size_t)BATCH * SEQ * EDIM;     // [8,1,2048,2048]

  k_qkv  <<<3 * 2 * 32, 32, 0, stream>>>(emb, Wq, Wk, Wv, qkv);
  k_s22  <<<1, 128, 0, stream>>>(qkv, s22);
  k_hist <<<BATCH, 256, 0, stream>>>(ids, mask, cnt, cntAll, cm);
  k_rows <<<BATCH, 32, 0, stream>>>(s22, cnt, cntAll, rowVal, w22);
  k_out22<<<BATCH * 2 * 32, 32, 0, stream>>>(w22, qkv, out22);
  k_wout <<<BATCH * SEQ, 128, 0, stream>>>(ids, out22, out);
  k_wattn<<<BATCH * (SEQ / 16), 256, 0, stream>>>(ids, cm, rowVal, attn);
}